// GCN_node_classification_57750130262575
// MI455X (gfx1250) — compile-verified
//
#include <hip/hip_runtime.h>
#include <hip/hip_bf16.h>

#define NN 50000
#define FH 128
#define NC 2
#define BN_EPS 1e-5f

typedef float v2f __attribute__((ext_vector_type(2)));
typedef float v8f __attribute__((ext_vector_type(8)));

// ---------------- BN0: h0 = (x - mean) * rsqrt(var+eps) * gamma + beta ----------------
__global__ __launch_bounds__(256) void bn0_kernel(
    const float* __restrict__ x, const float* __restrict__ gamma,
    const float* __restrict__ beta, const float* __restrict__ mean,
    const float* __restrict__ var, float* __restrict__ h0, long long total) {
  long long i = (long long)blockIdx.x * blockDim.x + threadIdx.x;
  if (i >= total) return;
  int f = (int)(i & (FH - 1));
  float inv = rsqrtf(var[f] + BN_EPS);
  h0[i] = (x[i] - mean[f]) * inv * gamma[f] + beta[f];
}

// ---------------- degree: atomic count of dst, then dinv = rsqrt(deg+1) ----------------
__global__ __launch_bounds__(256) void degree_kernel(
    const int* __restrict__ dst, float* __restrict__ deg, int E) {
  int stride = gridDim.x * blockDim.x;
  for (int e = blockIdx.x * blockDim.x + threadIdx.x; e < E; e += stride)
    atomicAdd(&deg[dst[e]], 1.0f);
}

__global__ __launch_bounds__(256) void dinv_kernel(float* __restrict__ dinv, int n) {
  int i = blockIdx.x * blockDim.x + threadIdx.x;
  if (i < n) dinv[i] = rsqrtf(dinv[i] + 1.0f);  // +1 self loop; deg>=1 always
}

// ---------------- GEMM1: xw1[N,128] = h0[N,128] @ W1[128,128] via f32 WMMA ------------
// Block = 256 threads = 8 waves. Block owns one 16-row tile; wave w owns cols [16w,16w+16).
// A 16x4 f32 layout : lane l, vgpr j -> A[l%16][2*(l/16)+j]
// B 4x16  f32 layout : lane l, vgpr j -> B[2*(l/16)+j][l%16]
// C/D 16x16 f32      : lane l, vgpr r -> D[r + 8*(l/16)][l%16]
__global__ __launch_bounds__(256) void gemm1_wmma(
    const float* __restrict__ A, const float* __restrict__ W,
    float* __restrict__ Out, int nrows) {
  __shared__ float As[16 * FH];  // 8 KB tile of A
  const int row0 = blockIdx.x * 16;
  const int t = threadIdx.x;
  for (int i = t; i < 16 * FH; i += 256) {
    int r = i >> 7, c = i & (FH - 1);
    int rr = row0 + r; if (rr >= nrows) rr = nrows - 1;   // N is a multiple of 16; safety clamp
    As[i] = A[(long long)rr * FH + c];
  }
  __syncthreads();

  const int wave = t >> 5;
  const int lane = t & 31;
  const int col0 = wave * 16;
  const int mn   = lane & 15;    // m for A, n for B, col for D
  const int half = lane >> 4;    // 0: K pair {0,1}; 1: K pair {2,3}

  v8f acc = {};
  #pragma unroll
  for (int kk = 0; kk < FH; kk += 4) {
    const int ka = kk + half * 2;
    v2f a, b;
    a[0] = As[mn * FH + ka];
    a[1] = As[mn * FH + ka + 1];
    b[0] = W[(long long)ka * FH + col0 + mn];
    b[1] = W[(long long)(ka + 1) * FH + col0 + mn];
    acc = __builtin_amdgcn_wmma_f32_16x16x4_f32(
        /*neg_a=*/false, a, /*neg_b=*/false, b,
        /*c_mod=*/(short)0, acc, /*reuse_a=*/false, /*reuse_b=*/false);
  }
  #pragma unroll
  for (int r = 0; r < 8; ++r) {
    int rr = row0 + r + 8 * half;
    if (rr < nrows) Out[(long long)rr * FH + col0 + mn] = acc[r];
  }
}

// ---------------- Edge scatter 1: one wave per edge, float4 per lane ------------------
__global__ __launch_bounds__(256) void aggregate1_kernel(
    const float* __restrict__ xw, const int* __restrict__ src,
    const int* __restrict__ dst, const float* __restrict__ dinv,
    float* __restrict__ agg, int E) {
  long long gid = (long long)blockIdx.x * blockDim.x + threadIdx.x;
  long long wid = gid >> 5;
  int lane = threadIdx.x & 31;
  long long nwaves = ((long long)gridDim.x * blockDim.x) >> 5;
  for (long long e = wid; e < E; e += nwaves) {
    int s = src[e], d = dst[e];
    float w = dinv[s] * dinv[d];
    float4 v = ((const float4*)(xw + (long long)s * FH))[lane];  // global_load_b128
    float* out = agg + (long long)d * FH + lane * 4;
    atomicAdd(out + 0, v.x * w);
    atomicAdd(out + 1, v.y * w);
    atomicAdd(out + 2, v.z * w);
    atomicAdd(out + 3, v.w * w);
  }
}

// ---------------- finalize1: += self-loop + b1, BN1, ReLU (in place -> h1) ------------
__global__ __launch_bounds__(256) void finalize1_kernel(
    float* __restrict__ agg /* becomes h1 */, const float* __restrict__ xw1,
    const float* __restrict__ dinv, const float* __restrict__ b1,
    const float* __restrict__ gamma, const float* __restrict__ beta,
    const float* __restrict__ mean, const float* __restrict__ var, long long total) {
  long long i = (long long)blockIdx.x * blockDim.x + threadIdx.x;
  if (i >= total) return;
  int n = (int)(i >> 7), f = (int)(i & (FH - 1));
  float di = dinv[n];
  float v = agg[i] + xw1[i] * di * di + b1[f];
  v = (v - mean[f]) * rsqrtf(var[f] + BN_EPS) * gamma[f] + beta[f];
  agg[i] = fmaxf(v, 0.0f);
}

// ---------------- GEMM2: xw2[N,2] = h1[N,128] @ W2[128,2] (tiny; VALU) ----------------
__global__ __launch_bounds__(256) void gemm2_kernel(
    const float* __restrict__ h1, const float* __restrict__ W2,
    float* __restrict__ xw2, int n) {
  __shared__ float w[FH * NC];
  int t = threadIdx.x;
  if (t < FH * NC) w[t] = W2[t];
  __syncthreads();
  int node = blockIdx.x * blockDim.x + t;
  if (node >= n) return;
  const float4* row = (const float4*)(h1 + (long long)node * FH);
  float a0 = 0.f, a1 = 0.f;
  #pragma unroll
  for (int k4 = 0; k4 < FH / 4; ++k4) {
    float4 v = row[k4];
    int k = k4 * 4;
    a0 += v.x * w[(k + 0) * 2] + v.y * w[(k + 1) * 2] + v.z * w[(k + 2) * 2] + v.w * w[(k + 3) * 2];
    a1 += v.x * w[(k + 0) * 2 + 1] + v.y * w[(k + 1) * 2 + 1] + v.z * w[(k + 2) * 2 + 1] + v.w * w[(k + 3) * 2 + 1];
  }
  xw2[(long long)node * 2 + 0] = a0;
  xw2[(long long)node * 2 + 1] = a1;
}

// ---------------- Edge scatter 2: one thread per edge (2 feats) -----------------------
__global__ __launch_bounds__(256) void aggregate2_kernel(
    const float* __restrict__ xw2, const int* __restrict__ src,
    const int* __restrict__ dst, const float* __restrict__ dinv,
    float* __restrict__ agg2, int E) {
  int stride = gridDim.x * blockDim.x;
  for (int e = blockIdx.x * blockDim.x + threadIdx.x; e < E; e += stride) {
    int s = src[e], d = dst[e];
    float w = dinv[s] * dinv[d];
    float2 v = ((const float2*)xw2)[s];
    atomicAdd(&agg2[(long long)d * 2 + 0], v.x * w);
    atomicAdd(&agg2[(long long)d * 2 + 1], v.y * w);
  }
}

// ---------------- finalize2: out = agg2 + self-loop + b2 ------------------------------
__global__ __launch_bounds__(256) void finalize2_kernel(
    const float* __restrict__ agg2, const float* __restrict__ xw2,
    const float* __restrict__ dinv, const float* __restrict__ b2,
    float* __restrict__ out, int n) {
  int i = blockIdx.x * blockDim.x + threadIdx.x;
  if (i >= n * NC) return;
  int node = i >> 1, c = i & 1;
  float di = dinv[node];
  out[i] = agg2[i] + xw2[i] * di * di + b2[c];
}

extern "C" void kernel_launch(void* const* d_in, const int* in_sizes, int n_in,
                              void* d_out, int out_size, void* d_ws, size_t ws_size,
                              hipStream_t stream) {
  const float* x       = (const float*)d_in[0];
  const int*   edges   = (const int*)d_in[1];
  const float* bn0_g   = (const float*)d_in[2];
  const float* bn0_b   = (const float*)d_in[3];
  const float* bn0_m   = (const float*)d_in[4];
  const float* bn0_v   = (const float*)d_in[5];
  const float* W1      = (const float*)d_in[6];
  const float* b1      = (const float*)d_in[7];
  const float* bn1_g   = (const float*)d_in[8];
  const float* bn1_b   = (const float*)d_in[9];
  const float* bn1_m   = (const float*)d_in[10];
  const float* bn1_v   = (const float*)d_in[11];
  const float* W2      = (const float*)d_in[12];
  const float* b2      = (const float*)d_in[13];

  const int n = in_sizes[0] / FH;      // 50000
  const int E = in_sizes[1] / 2;       // 1.6M
  const int* src = edges;
  const int* dst = edges + E;

  // Workspace layout (floats): bufA [N*128] (h0 -> agg1 -> h1), bufB [N*128] (xw1),
  // dinv [N], xw2 [N*2], agg2 [N*2]
  float* bufA = (float*)d_ws;
  float* bufB = bufA + (long long)n * FH;
  float* dinv = bufB + (long long)n * FH;
  float* xw2  = dinv + n;
  float* agg2 = xw2 + (long long)n * NC;

  const long long totNF = (long long)n * FH;

  // ---- stage 0: BN0 + degrees ----
  hipMemsetAsync(dinv, 0, (size_t)n * sizeof(float), stream);
  bn0_kernel<<<(int)((totNF + 255) / 256), 256, 0, stream>>>(x, bn0_g, bn0_b, bn0_m, bn0_v, bufA, totNF);
  degree_kernel<<<2048, 256, 0, stream>>>(dst, dinv, E);
  dinv_kernel<<<(n + 255) / 256, 256, 0, stream>>>(dinv, n);

  // ---- stage 1: GEMM1 (WMMA f32), scatter, finalize ----
  gemm1_wmma<<<(n + 15) / 16, 256, 0, stream>>>(bufA, W1, bufB, n);
  hipMemsetAsync(bufA, 0, (size_t)totNF * sizeof(float), stream);   // bufA becomes agg1
  aggregate1_kernel<<<4096, 256, 0, stream>>>(bufB, src, dst, dinv, bufA, E);
  finalize1_kernel<<<(int)((totNF + 255) / 256), 256, 0, stream>>>(
      bufA, bufB, dinv, b1, bn1_g, bn1_b, bn1_m, bn1_v, totNF);      // bufA becomes h1

  // ---- stage 2: GEMM2, scatter, finalize ----
  gemm2_kernel<<<(n + 255) / 256, 256, 0, stream>>>(bufA, W2, xw2, n);
  hipMemsetAsync(agg2, 0, (size_t)n * NC * sizeof(float), stream);
  aggregate2_kernel<<<2048, 256, 0, stream>>>(xw2, src, dst, dinv, agg2, E);
  finalize2_kernel<<<(n * NC + 255) / 256, 256, 0, stream>>>(agg2, xw2, dinv, b2, (float*)d_out, n);
}